// Net_56169582297455
// MI455X (gfx1250) — compile-verified
//
#include <hip/hip_runtime.h>
#include <hip/hip_bf16.h>
#include <stdint.h>

// Farthest-point sampling, npoint=2, on MI455X (gfx1250, wave32).
// Pass 1: argmax over y.  Pass 2: argmax of squared distance to seed point.
// Bandwidth-bound (~512 MB total traffic -> ~22us at 23.3 TB/s), so the kernel
// is built around CDNA5 data movement: 4-deep async global->LDS B128 ring
// (ASYNCcnt partial waits), non-temporal hints for the read-once streams,
// global_prefetch_b8, and wave32 shuffle reductions.

#define TPB 256              // threads per partial-reduction block (8 wave32s)
#define CPB 128              // chunks (blocks) per batch
#define TILE (TPB * 4)       // 1024 floats per tile per stream
#define PIPE 4               // async pipeline depth (ring of 4 LDS buffers)
#define NEG_INF (-3.402823466e38f)

__device__ __forceinline__ void amaxCombine(float& bv, int& bi, float v, int i) {
  // first-occurrence argmax semantics: prefer lower index on ties
  if (v > bv || (v == bv && i < bi)) { bv = v; bi = i; }
}

template <int BDIM>
__device__ __forceinline__ void blockArgmax(float& bv, int& bi, float* swv, int* swi) {
  #pragma unroll
  for (int off = 16; off > 0; off >>= 1) {      // wave32 reduction
    float ov = __shfl_down(bv, off, 32);
    int   oi = __shfl_down(bi, off, 32);
    amaxCombine(bv, bi, ov, oi);
  }
  const int lane = threadIdx.x & 31;
  const int wave = threadIdx.x >> 5;
  constexpr int NW = BDIM / 32;
  if (lane == 0) { swv[wave] = bv; swi[wave] = bi; }
  __syncthreads();
  if (wave == 0) {
    bv = (lane < NW) ? swv[lane] : NEG_INF;
    bi = (lane < NW) ? swi[lane] : 0x7fffffff;
    #pragma unroll
    for (int off = 16; off > 0; off >>= 1) {
      float ov = __shfl_down(bv, off, 32);
      int   oi = __shfl_down(bi, off, 32);
      amaxCombine(bv, bi, ov, oi);
    }
  }
}

// ---------------- Pass 1: per-chunk argmax over y (direct loads + prefetch) --

__global__ __launch_bounds__(TPB)
void fps_ymax_partials(const float* __restrict__ xyz,
                       float* __restrict__ pvals, int* __restrict__ pidx,
                       int N, int epb) {
  const int b = blockIdx.y;
  const int chunk = blockIdx.x;
  const int t = threadIdx.x;
  const float* __restrict__ Yb = xyz + (size_t)b * 3u * (size_t)N + (size_t)N;
  const int start = chunk * epb;
  const int end = (start + epb < N) ? (start + epb) : N;
  int span = end - start; if (span < 0) span = 0;
  const int nFull = span / TILE;

  float bv = NEG_INF; int bi = 0x7fffffff;
  for (int tile = 0; tile < nFull; ++tile) {
    const int n0 = start + tile * TILE + t * 4;
    __builtin_prefetch(Yb + n0 + 4 * TILE, 0, 0);   // global_prefetch_b8, 16KB ahead
    const float4 y = *reinterpret_cast<const float4*>(Yb + n0);
    if (y.x > bv) { bv = y.x; bi = n0;     }
    if (y.y > bv) { bv = y.y; bi = n0 + 1; }
    if (y.z > bv) { bv = y.z; bi = n0 + 2; }
    if (y.w > bv) { bv = y.w; bi = n0 + 3; }
  }
  for (int n = start + nFull * TILE + t; n < end; n += TPB) {
    const float v = Yb[n];
    if (v > bv || (v == bv && n < bi)) { bv = v; bi = n; }
  }
  __shared__ float swv[TPB / 32];
  __shared__ int   swi[TPB / 32];
  blockArgmax<TPB>(bv, bi, swv, swi);
  if (t == 0) { pvals[b * CPB + chunk] = bv; pidx[b * CPB + chunk] = bi; }
}

// ---------------- Finalize pass 1: pick seed index, gather centroid ---------

__global__ __launch_bounds__(CPB)
void fps_finalize1(const float* __restrict__ xyz,
                   const float* __restrict__ pvals, const int* __restrict__ pidx,
                   int* __restrict__ out, float* __restrict__ cent, int N) {
  const int b = blockIdx.x;
  const int t = threadIdx.x;
  float bv = pvals[b * CPB + t];
  int   bi = pidx[b * CPB + t];
  __shared__ float swv[CPB / 32];
  __shared__ int   swi[CPB / 32];
  blockArgmax<CPB>(bv, bi, swv, swi);
  if (t == 0) {
    out[2 * b] = bi;                                  // centroids[b][0]
    const float* Xb = xyz + (size_t)b * 3u * (size_t)N;
    cent[b * 4 + 0] = Xb[bi];
    cent[b * 4 + 1] = Xb[(size_t)N + bi];
    cent[b * 4 + 2] = Xb[2u * (size_t)N + bi];
  }
}

// ---------------- Pass 2: squared-distance argmax, 4-deep async pipeline ----

__global__ __launch_bounds__(TPB)
void fps_dist_partials(const float* __restrict__ xyz, const float* __restrict__ cent,
                       float* __restrict__ pvals, int* __restrict__ pidx,
                       int N, int epb) {
  const int b = blockIdx.y;
  const int chunk = blockIdx.x;
  const int t = threadIdx.x;
  const float* __restrict__ Xb = xyz + (size_t)b * 3u * (size_t)N;
  const float* __restrict__ Yb = Xb + (size_t)N;
  const float* __restrict__ Zb = Xb + 2u * (size_t)N;
  const float cx = cent[b * 4 + 0];
  const float cy = cent[b * 4 + 1];
  const float cz = cent[b * 4 + 2];

  const int start = chunk * epb;
  const int end = (start + epb < N) ? (start + epb) : N;
  int span = end - start; if (span < 0) span = 0;
  const int nFull = span / TILE;

  // 4-deep ring of staging buffers: [ring][x|y|z][thread] -> 48KB LDS
  __shared__ float4 stage[PIPE][3][TPB];
  // Low 32 bits of a generic LDS pointer == byte offset within the WG's LDS.
  uint32_t ldsA[PIPE][3];
  #pragma unroll
  for (int p = 0; p < PIPE; ++p)
    #pragma unroll
    for (int k = 0; k < 3; ++k)
      ldsA[p][k] = (uint32_t)(uintptr_t)&stage[p][k][t];

  // Each lane asynchronously stages its own 16B per stream (GVS addressing):
  // copies 16B from global saddr plus the byte offset in vaddr to the LDS address in vdst; tracked by this wave's ASYNCcnt.
  // NT hint: x/y/z in this pass are read exactly once -> don't thrash L2.
  auto issue = [&](int p, int elemBase) {
    const uint32_t off = (uint32_t)((elemBase + t * 4) * 4);
    asm volatile("global_load_async_to_lds_b128 %0, %1, %2 offset:0 th:TH_LOAD_NT"
                 :: "v"(ldsA[p][0]), "v"(off), "s"(Xb) : "memory");
    asm volatile("global_load_async_to_lds_b128 %0, %1, %2 offset:0 th:TH_LOAD_NT"
                 :: "v"(ldsA[p][1]), "v"(off), "s"(Yb) : "memory");
    asm volatile("global_load_async_to_lds_b128 %0, %1, %2 offset:0 th:TH_LOAD_NT"
                 :: "v"(ldsA[p][2]), "v"(off), "s"(Zb) : "memory");
  };

  // Prologue: fill the ring (up to PIPE tiles, 3 asyncs each).
  const int nPro = (nFull < PIPE) ? nFull : PIPE;
  for (int p = 0; p < nPro; ++p) issue(p, start + p * TILE);

  float bv = NEG_INF; int bi = 0x7fffffff;
  for (int tile = 0; tile < nFull; ++tile) {
    const int buf = tile & (PIPE - 1);
    const int curBase = start + tile * TILE;
    // Tiles still outstanding beyond the current one (ASYNCcnt retires in
    // order, so a partial wait pins exactly the newest 3*rem transfers).
    const int rem = nFull - 1 - tile;
    if      (rem >= 3) asm volatile("s_wait_asynccnt 0x9" ::: "memory");
    else if (rem == 2) asm volatile("s_wait_asynccnt 0x6" ::: "memory");
    else if (rem == 1) asm volatile("s_wait_asynccnt 0x3" ::: "memory");
    else               asm volatile("s_wait_asynccnt 0x0" ::: "memory");

    // Each thread reads back exactly the LDS slot it staged (ds_load_b128);
    // no cross-lane sharing, so no workgroup barrier in the hot loop.
    const float4 x = stage[buf][0][t];
    const float4 y = stage[buf][1][t];
    const float4 z = stage[buf][2][t];

    // Re-issue into this buffer for tile+PIPE. The ds_loads above may still
    // be in flight (DScnt) and are unordered vs the async write engine, so
    // close the WAR window explicitly before overwriting the buffer.
    if (tile + PIPE < nFull) {
      asm volatile("s_wait_dscnt 0x0" ::: "memory");
      issue(buf, curBase + PIPE * TILE);
    }

    const int n0 = curBase + t * 4;
    float dx, dy, dz, d;
    dx = x.x - cx; dy = y.x - cy; dz = z.x - cz;
    d = fminf(dx * dx + dy * dy + dz * dz, 1e10f);
    if (d > bv) { bv = d; bi = n0; }
    dx = x.y - cx; dy = y.y - cy; dz = z.y - cz;
    d = fminf(dx * dx + dy * dy + dz * dz, 1e10f);
    if (d > bv) { bv = d; bi = n0 + 1; }
    dx = x.z - cx; dy = y.z - cy; dz = z.z - cz;
    d = fminf(dx * dx + dy * dy + dz * dz, 1e10f);
    if (d > bv) { bv = d; bi = n0 + 2; }
    dx = x.w - cx; dy = y.w - cy; dz = z.w - cz;
    d = fminf(dx * dx + dy * dy + dz * dz, 1e10f);
    if (d > bv) { bv = d; bi = n0 + 3; }
  }
  // ragged tail: guarded scalar loads
  for (int n = start + nFull * TILE + t; n < end; n += TPB) {
    const float dx = Xb[n] - cx, dy = Yb[n] - cy, dz = Zb[n] - cz;
    const float d = fminf(dx * dx + dy * dy + dz * dz, 1e10f);
    if (d > bv || (d == bv && n < bi)) { bv = d; bi = n; }
  }

  __shared__ float swv[TPB / 32];
  __shared__ int   swi[TPB / 32];
  blockArgmax<TPB>(bv, bi, swv, swi);
  if (t == 0) { pvals[b * CPB + chunk] = bv; pidx[b * CPB + chunk] = bi; }
}

// ---------------- Finalize pass 2 -------------------------------------------

__global__ __launch_bounds__(CPB)
void fps_finalize2(const float* __restrict__ pvals, const int* __restrict__ pidx,
                   int* __restrict__ out) {
  const int b = blockIdx.x;
  const int t = threadIdx.x;
  float bv = pvals[b * CPB + t];
  int   bi = pidx[b * CPB + t];
  __shared__ float swv[CPB / 32];
  __shared__ int   swi[CPB / 32];
  blockArgmax<CPB>(bv, bi, swv, swi);
  if (t == 0) out[2 * b + 1] = bi;                    // centroids[b][1]
}

// ---------------- Host launch ----------------------------------------------

extern "C" void kernel_launch(void* const* d_in, const int* in_sizes, int n_in,
                              void* d_out, int out_size, void* d_ws, size_t ws_size,
                              hipStream_t stream) {
  const float* xyz = (const float*)d_in[0];   // [1, B, 3, N] f32
  const int B = out_size / 2;                 // output is [B, 2] int32 indices
  const int N = in_sizes[0] / (3 * B);
  // chunk size per block, rounded to a multiple of 4 floats (16B alignment)
  const int epb = (((N + CPB - 1) / CPB) + 3) & ~3;

  float* pvals = (float*)d_ws;                       // B*CPB partial maxima
  int*   pidx  = (int*)(pvals + (size_t)B * CPB);    // B*CPB partial indices
  float* cent  = (float*)(pidx + (size_t)B * CPB);   // B*4 centroid coords
  int*   out   = (int*)d_out;

  dim3 gPart(CPB, B);
  dim3 gFin(B);
  fps_ymax_partials<<<gPart, TPB, 0, stream>>>(xyz, pvals, pidx, N, epb);
  fps_finalize1<<<gFin, CPB, 0, stream>>>(xyz, pvals, pidx, out, cent, N);
  fps_dist_partials<<<gPart, TPB, 0, stream>>>(xyz, cent, pvals, pidx, N, epb);
  fps_finalize2<<<gFin, CPB, 0, stream>>>(pvals, pidx, out);
}